// Raindrop_15401752723925
// MI455X (gfx1250) — compile-verified
//
#include <hip/hip_runtime.h>
#include <hip/hip_bf16.h>
#include <stdint.h>

// ---------------- dims ----------------
constexpr int Bb  = 64;
constexpr int Tt  = 128;
constexpr int Ss  = 64;
constexpr int Oo  = 4;
constexpr int Ee  = 64;
constexpr int Pp  = 16;
constexpr int Aa  = 10;
constexpr int OUTn = 128;
constexpr int Dcat = Aa + Pp;   // 26
constexpr int DH   = Ee + Pp;   // 80

typedef __attribute__((ext_vector_type(2))) float v2f;
typedef __attribute__((ext_vector_type(8))) float v8f;
typedef __attribute__((ext_vector_type(4))) unsigned int u32x4;
typedef __attribute__((ext_vector_type(4))) int i32x4;
typedef __attribute__((ext_vector_type(8))) int i32x8;

#if defined(__gfx1250__) && __has_builtin(__builtin_amdgcn_tensor_load_to_lds)
#define HAVE_TDM 1
#else
#define HAVE_TDM 0
#endif

#if HAVE_TDM
// ---------------------------------------------------------------------------
// Tensor Data Mover: global -> LDS tile load (D# per CDNA5 ISA ch.8).
// tile0 elements contiguous per row, tile1 rows (0 => 1-D), stride0 in
// elements, optional LDS padding: after every 2^(icode+1) DWORDs stored,
// skip (acode+1) DWORDs (reproduces padded LDS row strides).
// ---------------------------------------------------------------------------
__device__ inline void tdm_load_to_lds(unsigned lds_off, const void* gptr,
                                       unsigned tile0, unsigned tile1,
                                       unsigned td0, unsigned td1,
                                       unsigned long long stride0,
                                       unsigned pad_en, unsigned icode,
                                       unsigned acode) {
    const unsigned long long ga = (unsigned long long)(uintptr_t)gptr;
    u32x4 g0;
    g0.x = 1u;                                          // count=1, user D#
    g0.y = lds_off;                                     // lds_addr (bytes)
    g0.z = (unsigned)(ga & 0xffffffffu);                // global_addr[31:0]
    g0.w = (unsigned)((ga >> 32) & 0x01ffffffu) | (2u << 30); // [56:32] | type=2

    unsigned d0 = (2u << 16);                           // data_size = 4 bytes
    if (pad_en) d0 |= (1u << 20) | (icode << 22) | (acode << 25);
    i32x8 g1;
    g1[0] = (int)d0;                                    // mask=0 (not in cluster)
    g1[1] = (int)((td0 & 0xffffu) << 16);               // tensor_dim0 lo16
    g1[2] = (int)(((td0 >> 16) & 0xffffu) | ((td1 & 0xffffu) << 16));
    g1[3] = (int)(((td1 >> 16) & 0xffffu) | ((tile0 & 0xffffu) << 16));
    g1[4] = (int)(tile1 & 0xffffu);                     // tile_dim1 (tile_dim2=0)
    g1[5] = (int)(stride0 & 0xffffffffu);               // dim0 stride lo32
    g1[6] = (int)((stride0 >> 32) & 0xffffu);           // dim0 stride hi16
    g1[7] = 0;

    const i32x4 z4 = {0, 0, 0, 0};
#if __clang_major__ >= 23
    const i32x8 z8 = {0, 0, 0, 0, 0, 0, 0, 0};
    __builtin_amdgcn_tensor_load_to_lds(g0, g1, z4, z4, z8, 0);
#else
    __builtin_amdgcn_tensor_load_to_lds(g0, g1, z4, z4, 0);
#endif
}
#endif // HAVE_TDM

// ---------------------------------------------------------------------------
// kernel 0: bidir = bidir_weights @ bidir_weights^T   (64x64, tiny)
// ---------------------------------------------------------------------------
__global__ __launch_bounds__(256) void k0_bidir(const float* __restrict__ bw,
                                                float* __restrict__ bidirBuf) {
    for (int idx = threadIdx.x; idx < Ss * Ss; idx += 256) {
        int i = idx >> 6, j = idx & 63;
        float acc = 0.f;
        #pragma unroll 8
        for (int e = 0; e < Ee; ++e) acc += bw[i * Ee + e] * bw[j * Ee + e];
        bidirBuf[idx] = acc;
    }
}

// ---------------------------------------------------------------------------
// kernel 2b: maskInv[b][s] = 1 / sum_t mask[b][t][s]
// ---------------------------------------------------------------------------
__global__ __launch_bounds__(256) void k2b_maskinv(const float* __restrict__ mask,
                                                   float* __restrict__ maskInv) {
    int g = blockIdx.x * 256 + threadIdx.x;       // B*S = 4096
    int b = g >> 6, s = g & 63;
    float acc = 0.f;
    for (int t = 0; t < Tt; ++t) acc += mask[((size_t)b * Tt + t) * Ss + s];
    maskInv[g] = 1.f / acc;
}

// ---------------------------------------------------------------------------
// kernel 2a: alphaSum[b][i][j] = sum_t alpha[b][t][i][j]
// ---------------------------------------------------------------------------
__global__ __launch_bounds__(256) void k2a_alphasum(const float* __restrict__ alphaBuf,
                                                    float* __restrict__ alphaSum) {
    int g = blockIdx.x * 256 + threadIdx.x;       // B*S*S = 262144
    int b = g >> 12, ij = g & 4095;
    float acc = 0.f;
    for (int t = 0; t < Tt; ++t)
        acc += alphaBuf[((size_t)b * Tt + t) * (Ss * Ss) + ij];
    alphaSum[g] = acc;
}

// ---------------------------------------------------------------------------
// kernel 1: per (b,t): h0, h_map, alpha (spill), propagate1 -> h1 (spill)
// ---------------------------------------------------------------------------
__global__ __launch_bounds__(256) void k1_pass1(
    const float* __restrict__ x, const float* __restrict__ times,
    const float* __restrict__ mask, const float* __restrict__ obsW,
    const float* __restrict__ isaW, const float* __restrict__ Wrecv,
    const float* __restrict__ brecv, const float* __restrict__ bidirBuf,
    float* __restrict__ alphaBuf, float* __restrict__ h1Buf) {

    __shared__ float sh_h0[64][68];     // h0 (S x E)
    __shared__ float sh_wrecv[64][36];  // W_recv zero-padded to 32 cols
    __shared__ float sh_hmap[64][36];   // h_map zero-padded to 32 cols
    __shared__ float sh_attnT[32][68];  // attn_cat^T  [d][j], rows 26..31 zero
    __shared__ float sh_w[64][68];      // alpha, then w

    const int tid = threadIdx.x;
    const int bt  = blockIdx.x;                  // b*T + t
    const float tt = times[bt];
    const size_t btOff = (size_t)bt * (Ss * Ee); // == bt*S*S too (both 4096)

    // W_recv -> LDS (padded)
    for (int idx = tid; idx < 64 * 32; idx += 256) {
        int e = idx >> 5, d = idx & 31;
        sh_wrecv[e][d] = (d < Dcat) ? Wrecv[e * Dcat + d] : 0.f;
    }
    // attn_cat^T -> LDS  (inter-sensor weights then sinusoidal PE)
    const float nl = -logf(10000.f) / (float)Pp;
    for (int idx = tid; idx < 32 * 64; idx += 256) {
        int d = idx >> 6, j = idx & 63;
        float v = 0.f;
        if (d < Aa) {
            v = isaW[j * Aa + d];
        } else if (d < Dcat) {
            int d2 = d - Aa;
            float fr = expf(nl * (float)(2 * (d2 & 7)));
            v = (d2 < 8) ? sinf(tt * fr) : cosf(tt * fr);
        }
        sh_attnT[d][j] = v;
    }
    // h0 = relu(x @ obsW per sensor) * mask
    for (int idx = tid; idx < 64 * 64; idx += 256) {
        int s = idx >> 6, e = idx & 63;
        const float* xp = x + (size_t)bt * Ss * Oo + s * Oo;
        const float* wp = obsW + (size_t)s * Oo * Ee + e;
        float acc = xp[0] * wp[0] + xp[1] * wp[Ee] + xp[2] * wp[2 * Ee] + xp[3] * wp[3 * Ee];
        sh_h0[s][e] = fmaxf(acc, 0.f) * mask[(size_t)bt * Ss + s];
    }
    __syncthreads();

    const int wave = tid >> 5;
    const int lane = tid & 31;
    const int half = lane >> 4;
    const int l    = lane & 15;

    // h_map = h0 @ W_recv + b_recv   (M=64,N=32,K=64): 8 tiles, 1 per wave
    {
        const int tm = (wave >> 1) * 16;
        const int tn = (wave & 1) * 16;
        v8f acc = {};
        for (int k = 0; k < 64; k += 4) {
            v2f a, bf;
            a.x  = sh_h0[tm + l][k + 2 * half + 0];
            a.y  = sh_h0[tm + l][k + 2 * half + 1];
            bf.x = sh_wrecv[k + 2 * half + 0][tn + l];
            bf.y = sh_wrecv[k + 2 * half + 1][tn + l];
            acc = __builtin_amdgcn_wmma_f32_16x16x4_f32(false, a, false, bf,
                                                        (short)0, acc, false, false);
        }
        const int col = tn + l;
        const float bias = (col < Dcat) ? brecv[col] : 0.f;
        #pragma unroll
        for (int r = 0; r < 8; ++r)
            sh_hmap[tm + r + 8 * half][col] = acc[r] + bias;
    }
    __syncthreads();

    // alpha = relu(h_map @ attn_cat^T)  (M=64,N=64,K=32): 16 tiles, 2 per wave
    for (int tile = wave; tile < 16; tile += 8) {
        const int tm = (tile >> 2) * 16;
        const int tn = (tile & 3) * 16;
        v8f acc = {};
        for (int k = 0; k < 32; k += 4) {
            v2f a, bf;
            a.x  = sh_hmap[tm + l][k + 2 * half + 0];
            a.y  = sh_hmap[tm + l][k + 2 * half + 1];
            bf.x = sh_attnT[k + 2 * half + 0][tn + l];
            bf.y = sh_attnT[k + 2 * half + 1][tn + l];
            acc = __builtin_amdgcn_wmma_f32_16x16x4_f32(false, a, false, bf,
                                                        (short)0, acc, false, false);
        }
        #pragma unroll
        for (int r = 0; r < 8; ++r) {
            const int row = tm + r + 8 * half, col = tn + l;
            const float v = fmaxf(acc[r], 0.f);
            sh_w[row][col] = v;
            alphaBuf[btOff + row * 64 + col] = v;
        }
    }
    __syncthreads();

    // w = bidir * alpha  (adj1 == 1)
    for (int idx = tid; idx < 4096; idx += 256)
        sh_w[idx >> 6][idx & 63] *= bidirBuf[idx];
    __syncthreads();

    // h1 = relu(w @ h0)  (M=64,N=64,K=64): 16 tiles, 2 per wave
    for (int tile = wave; tile < 16; tile += 8) {
        const int tm = (tile >> 2) * 16;
        const int tn = (tile & 3) * 16;
        v8f acc = {};
        for (int k = 0; k < 64; k += 4) {
            v2f a, bf;
            a.x  = sh_w[tm + l][k + 2 * half + 0];
            a.y  = sh_w[tm + l][k + 2 * half + 1];
            bf.x = sh_h0[k + 2 * half + 0][tn + l];
            bf.y = sh_h0[k + 2 * half + 1][tn + l];
            acc = __builtin_amdgcn_wmma_f32_16x16x4_f32(false, a, false, bf,
                                                        (short)0, acc, false, false);
        }
        #pragma unroll
        for (int r = 0; r < 8; ++r)
            h1Buf[btOff + (tm + r + 8 * half) * 64 + tn + l] = fmaxf(acc[r], 0.f);
    }
}

// ---------------------------------------------------------------------------
// kernel 3: per (b,t): w = bidir*alpha*adj2 ; h2 = relu(w @ h1) (overwrites alpha)
// Tile loads done by the Tensor Data Mover when available.
// ---------------------------------------------------------------------------
__global__ __launch_bounds__(256) void k3_pass2(
    float* __restrict__ alphaBuf, const float* __restrict__ h1Buf,
    const float* __restrict__ bidirBuf, const float* __restrict__ alphaSum,
    const float* __restrict__ maskInv) {

    __shared__ float sh_h1[64][68];
    __shared__ float sh_w[64][68];

    const int tid = threadIdx.x;
    const int bt  = blockIdx.x;
    const int b   = bt >> 7;                     // /T
    const size_t off = (size_t)bt * 4096;

#if HAVE_TDM
    // TDM: 4096 contiguous f32 each, LDS-padded 4 DWORDs per 64 (stride 68)
    if (tid < 32) {
        tdm_load_to_lds((unsigned)(uintptr_t)&sh_h1[0][0], h1Buf + off,
                        4096u, 0u, 4096u, 1u, 4096ull, 1u, 5u, 3u);
        tdm_load_to_lds((unsigned)(uintptr_t)&sh_w[0][0], alphaBuf + off,
                        4096u, 0u, 4096u, 1u, 4096ull, 1u, 5u, 3u);
        __builtin_amdgcn_s_wait_tensorcnt((short)0);
    }
    __syncthreads();
    // w = alpha * bidir * adj2,  adj2[i][j] = alphaSum[b][i][j] * maskInv[b][i]
    for (int idx = tid; idx < 4096; idx += 256) {
        const int i = idx >> 6;
        sh_w[i][idx & 63] *= bidirBuf[idx] *
                             alphaSum[(size_t)b * 4096 + idx] * maskInv[b * 64 + i];
    }
#else
    for (int idx = tid; idx < 4096; idx += 256) {
        const int i = idx >> 6, j = idx & 63;
        sh_h1[i][j] = h1Buf[off + idx];
        sh_w[i][j] = alphaBuf[off + idx] * bidirBuf[idx] *
                     alphaSum[(size_t)b * 4096 + idx] * maskInv[b * 64 + i];
    }
#endif
    __syncthreads();

    const int wave = tid >> 5;
    const int lane = tid & 31;
    const int half = lane >> 4;
    const int l    = lane & 15;

    for (int tile = wave; tile < 16; tile += 8) {
        const int tm = (tile >> 2) * 16;
        const int tn = (tile & 3) * 16;
        v8f acc = {};
        for (int k = 0; k < 64; k += 4) {
            v2f a, bf;
            a.x  = sh_w[tm + l][k + 2 * half + 0];
            a.y  = sh_w[tm + l][k + 2 * half + 1];
            bf.x = sh_h1[k + 2 * half + 0][tn + l];
            bf.y = sh_h1[k + 2 * half + 1][tn + l];
            acc = __builtin_amdgcn_wmma_f32_16x16x4_f32(false, a, false, bf,
                                                        (short)0, acc, false, false);
        }
        #pragma unroll
        for (int r = 0; r < 8; ++r)
            alphaBuf[off + (tm + r + 8 * half) * 64 + tn + l] = fmaxf(acc[r], 0.f);
    }
}

// ---------------------------------------------------------------------------
// kernel 4: per (b,s): H = [h2|pe]; Q,K; beta = QK^T*scale; W_s; softmax; out
// h2 columns fetched by TDM (2-D strided tile) when available.
// ---------------------------------------------------------------------------
__global__ __launch_bounds__(128) void k4_attn(
    const float* __restrict__ times, const float* __restrict__ h2Buf,
    const float* __restrict__ Wq, const float* __restrict__ bq,
    const float* __restrict__ Wk, const float* __restrict__ bk,
    const float* __restrict__ Wsc, const float* __restrict__ bsc,
    const float* __restrict__ Wemb, const float* __restrict__ bemb,
    float* __restrict__ out) {

    __shared__ float sh_H[128][84];    // T x (E+P), rows padded 64 -> 84
    __shared__ float sh_Q[128][20];    // a padded 10 -> 16
    __shared__ float sh_KT[16][132];   // K^T : [a][u]
    __shared__ float sh_beta[128][132];
    __shared__ float sh_red[128];
    __shared__ float sh_p[128];
    __shared__ float sh_r[80];

    const int tid = threadIdx.x;       // row t
    const int b = blockIdx.x >> 6;
    const int s = blockIdx.x & 63;

#if HAVE_TDM
    // TDM: 128 rows x 64 f32, row stride 4096 elements in global,
    // LDS-padded 20 DWORDs per 64 stored (row stride 84)
    if (tid < 32) {
        tdm_load_to_lds((unsigned)(uintptr_t)&sh_H[0][0],
                        h2Buf + (size_t)b * Tt * (Ss * Ee) + s * Ee,
                        64u, 128u, 1u << 20, 1u << 20, 4096ull, 1u, 5u, 19u);
    }
    // sinusoidal PE columns (pad region) while the DMA is in flight
    {
        const float tt = times[b * Tt + tid];
        const float nl = -logf(10000.f) / (float)Pp;
        #pragma unroll
        for (int d2 = 0; d2 < Pp; ++d2) {
            const float fr = expf(nl * (float)(2 * (d2 & 7)));
            sh_H[tid][Ee + d2] = (d2 < 8) ? sinf(tt * fr) : cosf(tt * fr);
        }
    }
    if (tid < 32) __builtin_amdgcn_s_wait_tensorcnt((short)0);
    __syncthreads();
#else
    {
        const float* hp = h2Buf + ((size_t)(b * Tt + tid)) * (Ss * Ee) + s * Ee;
        #pragma unroll 8
        for (int d = 0; d < Ee; ++d) sh_H[tid][d] = hp[d];
        const float tt = times[b * Tt + tid];
        const float nl = -logf(10000.f) / (float)Pp;
        #pragma unroll
        for (int d2 = 0; d2 < Pp; ++d2) {
            const float fr = expf(nl * (float)(2 * (d2 & 7)));
            sh_H[tid][Ee + d2] = (d2 < 8) ? sinf(tt * fr) : cosf(tt * fr);
        }
    }
    __syncthreads();
#endif

    // Q,K rows (a = 10, zero-padded to 16)
    {
        float q[16], kk[16];
        #pragma unroll
        for (int a = 0; a < 16; ++a) {
            q[a]  = (a < Aa) ? bq[a] : 0.f;
            kk[a] = (a < Aa) ? bk[a] : 0.f;
        }
        for (int d = 0; d < DH; ++d) {
            const float h = sh_H[tid][d];
            #pragma unroll
            for (int a = 0; a < Aa; ++a) {
                q[a]  += h * Wq[d * Aa + a];
                kk[a] += h * Wk[d * Aa + a];
            }
        }
        #pragma unroll
        for (int a = 0; a < 16; ++a) {
            sh_Q[tid][a]  = q[a];
            sh_KT[a][tid] = kk[a];
        }
    }
    __syncthreads();

    // beta = Q @ K^T * scale  (M=128,N=128,K=16): 64 tiles over 4 waves
    const int wave = tid >> 5;
    const int lane = tid & 31;
    const int half = lane >> 4;
    const int l    = lane & 15;
    const float scale = rsqrtf((float)DH);
    for (int tile = wave; tile < 64; tile += 4) {
        const int tm = (tile >> 3) * 16;
        const int tn = (tile & 7) * 16;
        v8f acc = {};
        for (int k = 0; k < 16; k += 4) {
            v2f a, bf;
            a.x  = sh_Q[tm + l][k + 2 * half + 0];
            a.y  = sh_Q[tm + l][k + 2 * half + 1];
            bf.x = sh_KT[k + 2 * half + 0][tn + l];
            bf.y = sh_KT[k + 2 * half + 1][tn + l];
            acc = __builtin_amdgcn_wmma_f32_16x16x4_f32(false, a, false, bf,
                                                        (short)0, acc, false, false);
        }
        #pragma unroll
        for (int r = 0; r < 8; ++r)
            sh_beta[tm + r + 8 * half][tn + l] = acc[r] * scale;
    }
    __syncthreads();

    // bscore[t] = beta[t] . W_s + b_s ; softmax over t (block-wide)
    float bs = bsc[0];
    for (int u = 0; u < Tt; ++u) bs += sh_beta[tid][u] * Wsc[u];
    sh_red[tid] = bs;
    __syncthreads();
    for (int o = 64; o > 0; o >>= 1) {
        if (tid < o) sh_red[tid] = fmaxf(sh_red[tid], sh_red[tid + o]);
        __syncthreads();
    }
    const float mx = sh_red[0];
    __syncthreads();
    const float ex = expf(bs - mx);
    sh_red[tid] = ex;
    __syncthreads();
    for (int o = 64; o > 0; o >>= 1) {
        if (tid < o) sh_red[tid] += sh_red[tid + o];
        __syncthreads();
    }
    sh_p[tid] = ex / sh_red[0];
    __syncthreads();

    // r[d] = sum_t p[t] * H[t][d]
    if (tid < DH) {
        float acc = 0.f;
        for (int t = 0; t < Tt; ++t) acc += sh_p[t] * sh_H[t][tid];
        sh_r[tid] = acc;
    }
    __syncthreads();

    // out = r @ W_emb + b_emb  (one output column per thread)
    {
        float acc = bemb[tid];
        for (int d = 0; d < DH; ++d) acc += sh_r[d] * Wemb[d * OUTn + tid];
        out[((size_t)(b * Ss + s)) * OUTn + tid] = acc;
    }
}

// ---------------------------------------------------------------------------
extern "C" void kernel_launch(void* const* d_in, const int* in_sizes, int n_in,
                              void* d_out, int out_size, void* d_ws, size_t ws_size,
                              hipStream_t stream) {
    (void)in_sizes; (void)n_in; (void)out_size; (void)ws_size;

    const float* x     = (const float*)d_in[0];
    const float* times = (const float*)d_in[1];
    const float* mask  = (const float*)d_in[2];
    const float* obsW  = (const float*)d_in[3];
    const float* isaW  = (const float*)d_in[4];
    const float* Wrecv = (const float*)d_in[5];
    const float* brecv = (const float*)d_in[6];
    const float* bw    = (const float*)d_in[7];
    const float* Wq    = (const float*)d_in[8];
    const float* bq    = (const float*)d_in[9];
    const float* Wk    = (const float*)d_in[10];
    const float* bk    = (const float*)d_in[11];
    const float* Wsc   = (const float*)d_in[12];
    const float* bsc   = (const float*)d_in[13];
    const float* Wemb  = (const float*)d_in[14];
    const float* bemb  = (const float*)d_in[15];
    float* out = (float*)d_out;

    float* ws       = (float*)d_ws;
    float* bidirBuf = ws;                               // 4096
    float* maskInv  = bidirBuf + 64 * 64;               // 4096
    float* alphaSum = maskInv + 64 * 64;                // 262144
    float* alphaBuf = alphaSum + 64 * 64 * 64;          // B*T*4096 (alpha, then h2)
    float* h1Buf    = alphaBuf + (size_t)Bb * Tt * 64 * 64;

    k0_bidir<<<1, 256, 0, stream>>>(bw, bidirBuf);
    k2b_maskinv<<<16, 256, 0, stream>>>(mask, maskInv);
    k1_pass1<<<Bb * Tt, 256, 0, stream>>>(x, times, mask, obsW, isaW, Wrecv, brecv,
                                          bidirBuf, alphaBuf, h1Buf);
    k2a_alphasum<<<1024, 256, 0, stream>>>(alphaBuf, alphaSum);
    k3_pass2<<<Bb * Tt, 256, 0, stream>>>(alphaBuf, h1Buf, bidirBuf, alphaSum, maskInv);
    k4_attn<<<Bb * Ss, 128, 0, stream>>>(times, alphaBuf /* h2 in place */, Wq, bq,
                                         Wk, bk, Wsc, bsc, Wemb, bemb, out);
}